// GINATT_Model_60653528154563
// MI455X (gfx1250) — compile-verified
//
#include <hip/hip_runtime.h>
#include <hip/hip_bf16.h>

typedef __attribute__((ext_vector_type(2))) float v2f;
typedef __attribute__((ext_vector_type(8))) float v8f;

#define N_NODES 100000
#define N_EDGES 600000
#define N_GRAPHS 2000
#define HID 128
#define BN_EPS 1e-5f

// ---------------------------------------------------------------------------
// Edge scatter-add: T[dst] += X[src]  (T pre-initialized to X, so T = h + agg)
// One wave (32 lanes) per edge; lanes stride channels -> 128B coalesced.
// ---------------------------------------------------------------------------
__global__ void scatter_add_kernel(const float* __restrict__ X,
                                   const int* __restrict__ src,
                                   const int* __restrict__ dst,
                                   float* __restrict__ T, int C) {
    int wave = (blockIdx.x * blockDim.x + threadIdx.x) >> 5;
    int lane = threadIdx.x & 31;
    if (wave >= N_EDGES) return;
    int s = src[wave];
    int d = dst[wave];
    const float* xs = X + (size_t)s * C;
    float*       td = T + (size_t)d * C;
    for (int c = lane; c < C; c += 32) {
        atomicAdd(&td[c], xs[c]);
    }
}

// ---------------------------------------------------------------------------
// Y[M x 128] = act(X[M x K] @ W[K x 128] + bias), act = ReLU if relu!=0.
// fp32 WMMA 16x16x4. One wave owns a 16(M) x 128(N) strip: 8 v8f accumulators.
// W is staged in LDS once per block as interleaved (k even, k odd) float2
// pairs so each B fragment is one ds_load_b64:
//   lw[(k>>1)*128 + col] = { W[k][col], W[k+1][col] }
// A 16x4 layout: lane(0-15)=row M, half=lane>>4 selects K pair {0,1}/{2,3}.
// C/D: VGPR v -> rows {v, v+8} selected by half. Row-block id is wave-uniform
// so EXEC is all-ones at every WMMA.
// ---------------------------------------------------------------------------
__global__ __launch_bounds__(256) void gemm_bias_relu_kernel(
        const float* __restrict__ X, const float* __restrict__ W,
        const float* __restrict__ bias, float* __restrict__ Y,
        int K, int relu) {
    __shared__ v2f lw[64 * HID];  // 64 KB: up to K=128 rows as 64 pair-rows

    // Cooperative LDS fill of the weight matrix (coalesced, once per block).
    int npairs = (K >> 1) * HID;
    for (int i = threadIdx.x; i < npairs; i += blockDim.x) {
        int kk  = i >> 7;    // pair-row = k/2
        int col = i & 127;
        v2f p;
        p.x = W[(2 * kk) * HID + col];
        p.y = W[(2 * kk + 1) * HID + col];
        lw[i] = p;
    }
    __syncthreads();

    int wave = blockIdx.x * (blockDim.x >> 5) + (threadIdx.x >> 5);
    int m0 = wave * 16;
    if (m0 >= N_NODES) return;  // wave-uniform tail exit (after barrier)

    int lane = threadIdx.x & 31;
    int half = lane >> 4;   // 0 or 1
    int idx  = lane & 15;   // row (A) / col (B,C,D)

    const float* xrow = X + (size_t)(m0 + idx) * K + 2 * half;

    v8f zero = {0.f, 0.f, 0.f, 0.f, 0.f, 0.f, 0.f, 0.f};
    v8f acc[8];
#pragma unroll
    for (int n = 0; n < 8; n++) acc[n] = zero;

    for (int k0 = 0; k0 < K; k0 += 16) {  // 4 k-steps per iteration
        v2f a[4];
#pragma unroll
        for (int i = 0; i < 4; i++) {
            const float* ap = xrow + k0 + 4 * i;
            a[i].x = ap[0];
            a[i].y = ap[1];
        }
#pragma unroll
        for (int i = 0; i < 4; i++) {
            const v2f* lb = &lw[(((k0 + 4 * i) >> 1) + half) * HID + idx];
            v2f b[8];
#pragma unroll
            for (int n = 0; n < 8; n++) b[n] = lb[n * 16];
#pragma unroll
            for (int n = 0; n < 8; n++)
                acc[n] = __builtin_amdgcn_wmma_f32_16x16x4_f32(
                    false, a[i], false, b[n], (short)0, acc[n], false, false);
        }
    }

#pragma unroll
    for (int n = 0; n < 8; n++) {
        int col = n * 16 + idx;
        float bv = bias[col];
#pragma unroll
        for (int v = 0; v < 8; v++) {
            int row = m0 + v + 8 * half;
            float val = acc[n][v] + bv;
            if (relu) val = fmaxf(val, 0.0f);
            Y[(size_t)row * HID + col] = val;
        }
    }
}

// ---------------------------------------------------------------------------
// BatchNorm statistics: per-channel sum and sum-of-squares over N_NODES rows.
// ---------------------------------------------------------------------------
__global__ void zero_stats_kernel(float* __restrict__ s) {
    s[threadIdx.x] = 0.0f;  // 256 threads: 128 sums + 128 sumsq
}

__global__ void bn_stats_kernel(const float* __restrict__ H,
                                float* __restrict__ sums,
                                float* __restrict__ sqs) {
    int c = threadIdx.x;  // 0..127
    float s = 0.f, q = 0.f;
    for (int r = blockIdx.x; r < N_NODES; r += gridDim.x) {
        float v = H[(size_t)r * HID + c];
        s += v;
        q += v * v;
    }
    atomicAdd(&sums[c], s);
    atomicAdd(&sqs[c], q);
}

__global__ void bn_apply_kernel(const float* __restrict__ H,
                                const float* __restrict__ sums,
                                const float* __restrict__ sqs,
                                const float* __restrict__ gamma,
                                const float* __restrict__ beta,
                                float* __restrict__ Xo) {
    int c = threadIdx.x;  // 0..127
    const float invN = 1.0f / (float)N_NODES;
    float mu  = sums[c] * invN;
    float var = sqs[c] * invN - mu * mu;
    float inv = rsqrtf(fmaxf(var, 0.0f) + BN_EPS);
    float g = gamma[c] * inv;
    float b = beta[c] - mu * g;
    for (int r = blockIdx.x; r < N_NODES; r += gridDim.x) {
        size_t i = (size_t)r * HID + c;
        Xo[i] = H[i] * g + b;
    }
}

// ---------------------------------------------------------------------------
// Global mean pool over sorted batch_ids: one block per graph, binary search
// its contiguous row range -> no atomics. Thread t covers concat channel t.
// ---------------------------------------------------------------------------
__device__ __forceinline__ int lower_bound_dev(const int* __restrict__ a,
                                               int n, int key) {
    int lo = 0, hi = n;
    while (lo < hi) {
        int mid = (lo + hi) >> 1;
        if (a[mid] < key) lo = mid + 1; else hi = mid;
    }
    return lo;
}

__global__ void pool_kernel(const float* __restrict__ x1,
                            const float* __restrict__ x2,
                            const float* __restrict__ x3,
                            const int* __restrict__ bids,
                            float* __restrict__ pooled) {
    int g = blockIdx.x;
    int lo = lower_bound_dev(bids, N_NODES, g);
    int hi = lower_bound_dev(bids, N_NODES, g + 1);
    int cnt = hi - lo;
    int t = threadIdx.x;  // 0..383
    const float* base = (t < 128) ? x1 : (t < 256) ? x2 : x3;
    int c = t & 127;
    float s = 0.f;
    for (int r = lo; r < hi; r++) s += base[(size_t)r * HID + c];
    pooled[(size_t)g * 384 + t] = s / fmaxf((float)cnt, 1.0f);
}

// ---------------------------------------------------------------------------
// out[2000 x 128] = pooled[2000 x 384] @ W_lin[384 x 128] + b_lin
// (0.2 GFLOP: latency-trivial, plain VALU is fine.)
// ---------------------------------------------------------------------------
__global__ void final_linear_kernel(const float* __restrict__ P,
                                    const float* __restrict__ W,
                                    const float* __restrict__ b,
                                    float* __restrict__ O) {
    int g = blockIdx.x;
    int j = threadIdx.x;  // 0..127
    const float* p = P + (size_t)g * 384;
    float acc = b[j];
    for (int c = 0; c < 384; c++) acc = fmaf(p[c], W[c * HID + j], acc);
    O[(size_t)g * HID + j] = acc;
}

// ---------------------------------------------------------------------------
extern "C" void kernel_launch(void* const* d_in, const int* in_sizes, int n_in,
                              void* d_out, int out_size, void* d_ws, size_t ws_size,
                              hipStream_t stream) {
    const float* x    = (const float*)d_in[0];
    const int*   ei   = (const int*)d_in[1];
    const int*   bids = (const int*)d_in[2];
    const int*   src  = ei;            // edge_index row 0
    const int*   dst  = ei + N_EDGES;  // edge_index row 1

    const size_t NF = (size_t)N_NODES * HID;  // 12.8M floats per buffer
    float* ws  = (float*)d_ws;
    float* x1  = ws;             // layer outputs persist for the concat
    float* x2  = x1 + NF;
    float* x3  = x2 + NF;
    float* tmp = x3 + NF;        // agg / h2 scratch
    float* y1  = tmp + NF;       // nn1 hidden
    float* stats  = y1 + NF;     // 256 floats: [sums | sumsq]
    float* pooled = stats + 256; // 2000*384 floats

    const float* xin[3]  = {x, x1, x2};
    float*       xout[3] = {x1, x2, x3};
    const int    Cin[3]  = {64, HID, HID};

    // 6250 wave-strips of 16 rows; 8 waves per 256-thread block.
    const int gemm_blocks = (6250 + 7) / 8;  // 782

    for (int l = 0; l < 3; l++) {
        const float* W1    = (const float*)d_in[3 + 6 * l];
        const float* b1    = (const float*)d_in[4 + 6 * l];
        const float* W2    = (const float*)d_in[5 + 6 * l];
        const float* b2    = (const float*)d_in[6 + 6 * l];
        const float* gamma = (const float*)d_in[7 + 6 * l];
        const float* beta  = (const float*)d_in[8 + 6 * l];
        const int C = Cin[l];

        // tmp = h, then tmp[dst] += h[src]  =>  tmp = h + segment_sum(h[src], dst)
        hipMemcpyAsync(tmp, xin[l], (size_t)N_NODES * C * sizeof(float),
                       hipMemcpyDeviceToDevice, stream);
        scatter_add_kernel<<<(N_EDGES * 32) / 256, 256, 0, stream>>>(
            xin[l], src, dst, tmp, C);

        // y1 = ReLU(tmp @ W1 + b1) ; h2 = ReLU(y1 @ W2 + b2)  (h2 reuses tmp)
        gemm_bias_relu_kernel<<<gemm_blocks, 256, 0, stream>>>(tmp, W1, b1, y1, C, 1);
        gemm_bias_relu_kernel<<<gemm_blocks, 256, 0, stream>>>(y1, W2, b2, tmp, HID, 1);

        // BatchNorm (train-mode, biased var)
        zero_stats_kernel<<<1, 256, 0, stream>>>(stats);
        bn_stats_kernel<<<512, 128, 0, stream>>>(tmp, stats, stats + 128);
        bn_apply_kernel<<<1024, 128, 0, stream>>>(tmp, stats, stats + 128,
                                                  gamma, beta, xout[l]);
    }

    pool_kernel<<<N_GRAPHS, 384, 0, stream>>>(x1, x2, x3, bids, pooled);
    final_linear_kernel<<<N_GRAPHS, 128, 0, stream>>>(
        pooled, (const float*)d_in[21], (const float*)d_in[22], (float*)d_out);
}